// FeatureExtractor_23536420782150
// MI455X (gfx1250) — compile-verified
//
#include <hip/hip_runtime.h>

// Problem constants (shape fixed by reference: (16, 3, 512, 512) fp32)
#define W            512
#define R_TOTAL      (16 * 3 * 512)         // 24576 rows total
#define TILE_ROWS    16
#define NTILES       (R_TOTAL / TILE_ROWS)  // 1536
#define NBINS        511
#define HIST_STRIDE  512                    // padded per-histogram stride
#define NHIST        4                      // gx_c, gy_c, gx_n, gy_n
#define HIST_INTS    (NHIST * HIST_STRIDE)  // 2048
#define THREADS      256
#define BLOCKS       768
#define RED_CHUNKS   16                     // reduce: chunks over BLOCKS
#define RED_PER      (BLOCKS / RED_CHUNKS)  // 48 block-hists per reduce thread

// ws layout: [0 .. 2048) int accumulator "acc"; [2048 ..) per-block dumps.
#define WS_NEEDED_PATH_A ((size_t)(1 + BLOCKS) * HIST_INTS * sizeof(int))

// ---------------------------------------------------------------------------
// Histogram bump: bin = clip(floor(g) + 255, 0, 510). Gradients of [0,255]
// data always lie in [-255,255], so the reference's range mask is always true.
// ---------------------------------------------------------------------------
__device__ __forceinline__ void bump(int* __restrict__ h, float g) {
    int b = __float2int_rd(g) + 255;
    b = b < 0 ? 0 : (b > 510 ? 510 : b);
    atomicAdd(&h[b], 1);   // ds_add_u32 (LDS atomic)
}

// ---------------------------------------------------------------------------
// Main kernel: per-block LDS-tiled gradient histograms.
//   - 16-row + 1 halo-row tiles DMA'd global->LDS with
//     global_load_async_to_lds_b128 (ASYNCcnt-tracked, 128b per lane, GVS).
//   - Gradients computed from LDS; counts in LDS-private histograms.
//   - merge_mode 0: plain b128 store of the block hist into its ws slot
//     (reduced later, no contention).  merge_mode 1: global atomic fallback.
// ---------------------------------------------------------------------------
__global__ __launch_bounds__(THREADS)
void hist_kernel(const float* __restrict__ clean,
                 const float* __restrict__ noisy,
                 int* __restrict__ ws,
                 int merge_mode) {
    __shared__ __align__(16) float tile[(TILE_ROWS + 1) * W];   // 34,816 B
    __shared__ __align__(16) int   hist[HIST_INTS];             //  8,192 B

    const int tid = threadIdx.x;

    for (int i = tid; i < HIST_INTS; i += THREADS) hist[i] = 0;

    // Low 32 bits of a generic pointer to LDS == 0-based LDS byte address.
    const unsigned tile_lds = (unsigned)(unsigned long long)(const void*)tile;

    for (int t = blockIdx.x; t < NTILES; t += gridDim.x) {
        const int gr0       = t * TILE_ROWS;
        const int rows_load = (R_TOTAL - gr0) < (TILE_ROWS + 1)
                                ? (R_TOTAL - gr0) : (TILE_ROWS + 1);
        const int n4        = rows_load * (W / 4);              // float4s
        const unsigned gbase = (unsigned)(gr0 * (W / 4)) * 16u; // byte offset

        for (int img = 0; img < 2; ++img) {
            __syncthreads();   // tile buffer free (previous readers done)

            const float* src = img ? noisy : clean;
            const unsigned long long sbase = (unsigned long long)src;

            // Async 128-bit global->LDS copies (EXEC-masked per lane).
            for (int i = tid; i < n4; i += THREADS) {
                unsigned lds_a = tile_lds + (unsigned)i * 16u;
                unsigned goff  = gbase    + (unsigned)i * 16u;
                asm volatile(
                    "global_load_async_to_lds_b128 %0, %1, %2"
                    :
                    : "v"(lds_a), "v"(goff), "s"(sbase)
                    : "memory");
            }
            asm volatile("s_wait_asynccnt 0" ::: "memory");
            __syncthreads();   // all waves' tile data visible in LDS

            int* hgx = &hist[(img * 2 + 0) * HIST_STRIDE];
            int* hgy = &hist[(img * 2 + 1) * HIST_STRIDE];

            // 16 rows x 128 float4 = 2048 float4s per tile; 8 per thread.
            for (int i4 = tid; i4 < TILE_ROWS * (W / 4); i4 += THREADS) {
                const int r  = i4 >> 7;
                const int c4 = i4 & 127;
                const float4 a = *(const float4*)&tile[r * W + c4 * 4];

                // gx: intra-row diffs (511 per row)
                bump(hgx, a.y - a.x);
                bump(hgx, a.z - a.y);
                bump(hgx, a.w - a.z);
                if (c4 < 127) {
                    float nxt = tile[r * W + c4 * 4 + 4];
                    bump(hgx, nxt - a.w);
                }

                // gy: inter-row diffs, masked across 512-row slice boundary
                if (((gr0 + r) & 511) != 511) {
                    const float4 b = *(const float4*)&tile[(r + 1) * W + c4 * 4];
                    bump(hgy, b.x - a.x);
                    bump(hgy, b.y - a.y);
                    bump(hgy, b.z - a.z);
                    bump(hgy, b.w - a.w);
                }
            }
        }
    }

    __syncthreads();
    if (merge_mode == 0) {
        // Contention-free: dump block hist (int4 stores) into its ws slot.
        int* dump = ws + HIST_INTS + (size_t)blockIdx.x * HIST_INTS;
        for (int i4 = tid; i4 < HIST_INTS / 4; i4 += THREADS) {
            *(int4*)&dump[i4 * 4] = *(const int4*)&hist[i4 * 4];
        }
    } else {
        // Fallback: atomic merge directly into acc.
        for (int i = tid; i < HIST_INTS; i += THREADS) {
            int v = hist[i];
            if (v != 0 && (i & (HIST_STRIDE - 1)) < NBINS)
                atomicAdd(&ws[i], v);
        }
    }
}

// ---------------------------------------------------------------------------
// Zero the 8 KB accumulator at the front of ws.
// ---------------------------------------------------------------------------
__global__ void zero_ws_kernel(int* __restrict__ ws) {
    int i = blockIdx.x * blockDim.x + threadIdx.x;
    if (i < HIST_INTS) ws[i] = 0;
}

// ---------------------------------------------------------------------------
// Path-A reduce: 512 int4-columns x RED_CHUNKS chunks; each thread sums
// RED_PER block-hists (coalesced int4 loads) and does 4 atomics into acc.
// Total atomics: 512*16*4 = 32K (vs 1.57M in the naive merge).
// ---------------------------------------------------------------------------
__global__ void reduce_kernel(const int* __restrict__ ws_dump,  // = ws + 2048
                              int* __restrict__ acc) {          // = ws
    int g = blockIdx.x * blockDim.x + threadIdx.x;
    if (g >= (HIST_INTS / 4) * RED_CHUNKS) return;
    const int i4    = g & (HIST_INTS / 4 - 1);   // int4 column, 0..511
    const int chunk = g >> 9;                    // 0..RED_CHUNKS-1
    int4 s = make_int4(0, 0, 0, 0);
    const int b0 = chunk * RED_PER;
    for (int j = 0; j < RED_PER; ++j) {
        const int4 v = *(const int4*)&ws_dump[(size_t)(b0 + j) * HIST_INTS + i4 * 4];
        s.x += v.x; s.y += v.y; s.z += v.z; s.w += v.w;
    }
    atomicAdd(&acc[i4 * 4 + 0], s.x);
    atomicAdd(&acc[i4 * 4 + 1], s.y);
    atomicAdd(&acc[i4 * 4 + 2], s.z);
    atomicAdd(&acc[i4 * 4 + 3], s.w);
}

// ---------------------------------------------------------------------------
// Convert acc counts -> float d_out (drops the pad bin; fully overwrites).
// ---------------------------------------------------------------------------
__global__ void convert_kernel(const int* __restrict__ acc,
                               float* __restrict__ out) {
    int i = blockIdx.x * blockDim.x + threadIdx.x;
    if (i < NHIST * NBINS) {
        int h = i / NBINS;
        int b = i - h * NBINS;
        out[i] = (float)acc[h * HIST_STRIDE + b];
    }
}

// ---------------------------------------------------------------------------
extern "C" void kernel_launch(void* const* d_in, const int* in_sizes, int n_in,
                              void* d_out, int out_size, void* d_ws, size_t ws_size,
                              hipStream_t stream) {
    const float* clean = (const float*)d_in[0];
    const float* noisy = (const float*)d_in[1];
    int*   ws  = (int*)d_ws;
    float* out = (float*)d_out;

    // Path choice is a pure function of ws_size (fixed per setup): deterministic.
    const int merge_mode = (ws_size >= WS_NEEDED_PATH_A) ? 0 : 1;

    zero_ws_kernel<<<(HIST_INTS + 255) / 256, 256, 0, stream>>>(ws);
    hist_kernel<<<BLOCKS, THREADS, 0, stream>>>(clean, noisy, ws, merge_mode);
    if (merge_mode == 0) {
        const int red_threads = (HIST_INTS / 4) * RED_CHUNKS;   // 8192
        reduce_kernel<<<(red_threads + 255) / 256, 256, 0, stream>>>(
            ws + HIST_INTS, ws);
    }
    convert_kernel<<<(NHIST * NBINS + 255) / 256, 256, 0, stream>>>(ws, out);
}